// LFE_86663850098728
// MI455X (gfx1250) — compile-verified
//
#include <hip/hip_runtime.h>

// Problem constants (match reference)
#define Bn    8
#define Nn    4096
#define CIN   64
#define Mn    1024
#define Kn    32
#define CCOV  64
#define COUT  128

typedef __attribute__((ext_vector_type(2))) float v2f;
typedef __attribute__((ext_vector_type(8))) float v8f;

// ---------------------------------------------------------------------------
// Wave32 helpers
// ---------------------------------------------------------------------------
__device__ __forceinline__ float wave_sum(float v) {
#pragma unroll
  for (int off = 16; off > 0; off >>= 1) v += __shfl_xor(v, off, 32);
  return v;
}
__device__ __forceinline__ float wave_max(float v) {
#pragma unroll
  for (int off = 16; off > 0; off >>= 1) v = fmaxf(v, __shfl_xor(v, off, 32));
  return v;
}

// ---------------------------------------------------------------------------
// Kernel 1: Farthest Point Sampling. One block per batch, 256 threads,
// 16 points per thread held in registers. Two barriers per iteration.
// ---------------------------------------------------------------------------
__global__ void fps_kernel(const float* __restrict__ xyz,
                           int* __restrict__ sidx,
                           float* __restrict__ xyz_new) {
  const int b   = blockIdx.x;
  const int tid = threadIdx.x;
  const float* pts = xyz + (size_t)b * Nn * 3;

  float px[16], py[16], pz[16], md[16];
#pragma unroll
  for (int i = 0; i < 16; ++i) {
    const int n = tid + i * 256;
    px[i] = pts[n * 3 + 0];
    py[i] = pts[n * 3 + 1];
    pz[i] = pts[n * 3 + 2];
    md[i] = 1e10f;
  }

  __shared__ float rv[8];
  __shared__ int   ri[8];
  __shared__ int   sFar;

  int far = 0;
  for (int m = 0; m < Mn; ++m) {
    // coords of current farthest point (uniform address -> broadcast load)
    const float fx = pts[far * 3 + 0];
    const float fy = pts[far * 3 + 1];
    const float fz = pts[far * 3 + 2];
    if (tid == 0) {
      sidx[b * Mn + m] = far;
      xyz_new[(b * Mn + m) * 3 + 0] = fx;
      xyz_new[(b * Mn + m) * 3 + 1] = fy;
      xyz_new[(b * Mn + m) * 3 + 2] = fz;
    }
    // update min-dists, local argmax
    float bv = -1.0f;
    int   bi = 0x7fffffff;
#pragma unroll
    for (int i = 0; i < 16; ++i) {
      const float dx = px[i] - fx, dy = py[i] - fy, dz = pz[i] - fz;
      const float d = dx * dx + dy * dy + dz * dz;
      const float nm = fminf(md[i], d);
      md[i] = nm;
      const int n = tid + i * 256;
      if (nm > bv) { bv = nm; bi = n; }   // i ascending => first-max kept
    }
    // wave argmax (prefer smaller index on ties, matching jnp.argmax)
#pragma unroll
    for (int off = 16; off > 0; off >>= 1) {
      const float ov = __shfl_xor(bv, off, 32);
      const int   oi = __shfl_xor(bi, off, 32);
      if (ov > bv || (ov == bv && oi < bi)) { bv = ov; bi = oi; }
    }
    if ((tid & 31) == 0) { rv[tid >> 5] = bv; ri[tid >> 5] = bi; }
    __syncthreads();
    if (tid < 32) {
      float v  = (tid < 8) ? rv[tid] : -1.0f;
      int   ix = (tid < 8) ? ri[tid] : 0x7fffffff;
#pragma unroll
      for (int off = 4; off > 0; off >>= 1) {
        const float ov = __shfl_xor(v, off, 32);
        const int   oi = __shfl_xor(ix, off, 32);
        if (ov > v || (ov == v && oi < ix)) { v = ov; ix = oi; }
      }
      if (tid == 0) sFar = ix;
    }
    __syncthreads();
    far = sFar;
  }
}

// ---------------------------------------------------------------------------
// Kernel 2: brute-force KNN (k=32 smallest of 4096). One thread per query,
// per-thread candidate list in LDS (stride 33 to avoid bank conflicts),
// point cloud streamed through LDS in 512-point tiles.
// ---------------------------------------------------------------------------
__global__ void knn_kernel(const float* __restrict__ xyz,
                           const float* __restrict__ xyz_new,
                           int* __restrict__ knn) {
  __shared__ float tile[512 * 3];
  __shared__ float sD[128 * 33];
  __shared__ int   sI[128 * 33];

  const int tid = threadIdx.x;
  const int q   = blockIdx.x * 128 + tid;   // M==1024 is a multiple of 128
  const int b   = q >> 10;                  // q / Mn
  const float* pb = xyz + (size_t)b * Nn * 3;

  const float qx = xyz_new[q * 3 + 0];
  const float qy = xyz_new[q * 3 + 1];
  const float qz = xyz_new[q * 3 + 2];

  const int base = tid * 33;
  float worst = 1e30f;
  int   wslot = 0;
#pragma unroll
  for (int s = 0; s < 32; ++s) sD[base + s] = 1e30f;

  for (int t = 0; t < Nn; t += 512) {
    __syncthreads();
    for (int j = tid; j < 512 * 3; j += 128) tile[j] = pb[t * 3 + j];
    __syncthreads();
    for (int nn = 0; nn < 512; ++nn) {
      const float dx = qx - tile[nn * 3 + 0];
      const float dy = qy - tile[nn * 3 + 1];
      const float dz = qz - tile[nn * 3 + 2];
      const float d = dx * dx + dy * dy + dz * dz;
      if (d < worst) {
        sD[base + wslot] = d;
        sI[base + wslot] = t + nn;
        float wd = -1.0f; int ws = 0;
#pragma unroll
        for (int s = 0; s < 32; ++s) {
          const float v = sD[base + s];
          if (v > wd) { wd = v; ws = s; }
        }
        worst = wd; wslot = ws;
      }
    }
  }
#pragma unroll
  for (int s = 0; s < 32; ++s) knn[q * Kn + s] = sI[base + s];
}

// ---------------------------------------------------------------------------
// Kernel 3: gather + max-pool over neighbors + 3x3 covariance.
// One wave per query (lane == neighbor k), butterfly reductions.
// Outputs: fmax_ws[q][c] (m-major) and cov9_ws[q][o].
// ---------------------------------------------------------------------------
__global__ void group_kernel(const float* __restrict__ f,
                             const float* __restrict__ xyz,
                             const int* __restrict__ knn,
                             float* __restrict__ cov9_ws,
                             float* __restrict__ fmax_ws) {
  const int lane = threadIdx.x & 31;
  const int q    = blockIdx.x * 8 + (threadIdx.x >> 5);
  const int b    = q >> 10;
  const int nid  = knn[q * Kn + lane];

  const float* pb = xyz + (size_t)b * Nn * 3;
  const float x = pb[nid * 3 + 0];
  const float y = pb[nid * 3 + 1];
  const float z = pb[nid * 3 + 2];

  const float inv = 1.0f / (float)Kn;
  const float mx = wave_sum(x) * inv;
  const float my = wave_sum(y) * inv;
  const float mz = wave_sum(z) * inv;
  const float cx = x - mx, cy = y - my, cz = z - mz;

  const float xx = wave_sum(cx * cx) * inv;
  const float xy = wave_sum(cx * cy) * inv;
  const float xz = wave_sum(cx * cz) * inv;
  const float yy = wave_sum(cy * cy) * inv;
  const float yz = wave_sum(cy * cz) * inv;
  const float zz = wave_sum(cz * cz) * inv;

  if (lane < 9) {
    float o;
    switch (lane) {
      case 0: o = xx; break; case 1: o = xy; break; case 2: o = xz; break;
      case 3: o = xy; break; case 4: o = yy; break; case 5: o = yz; break;
      case 6: o = xz; break; case 7: o = yz; break; default: o = zz; break;
    }
    cov9_ws[q * 9 + lane] = o;
  }

  const float* fb = f + (size_t)b * CIN * Nn;
#pragma unroll 4
  for (int c = 0; c < CIN; ++c) {
    float v = fb[(size_t)c * Nn + nid];
    v = wave_max(v);
    if (lane == (c & 31)) fmax_ws[(size_t)q * CIN + c] = v;
  }
}

// ---------------------------------------------------------------------------
// Kernel 4: fused mlp_cov (VALU, K=9) + mlp_f GEMM via V_WMMA_F32_16X16X4_F32.
// Block: 256 threads (8 waves), one batch x 64-point tile.
// fcatT [64 m][132 (padded 128 j)] staged in LDS; wave w computes out-channel
// strip [w*16, w*16+16) x 64 points = 4 accumulator tiles, one shared A-frag
// stream from global W_f (L2-resident).
// ---------------------------------------------------------------------------
__global__ void mlp_kernel(const float* __restrict__ Wf,
                           const float* __restrict__ bf,
                           const float* __restrict__ Wc,
                           const float* __restrict__ bc,
                           const float* __restrict__ cov9_ws,
                           const float* __restrict__ fmax_ws,
                           float* __restrict__ out) {
  __shared__ float fcatT[64 * 132];

  const int b     = blockIdx.x >> 4;          // 16 m-tiles per batch
  const int mBase = (blockIdx.x & 15) * 64;
  const int tid   = threadIdx.x;
  const int q0    = b * Mn + mBase;

  // ---- stage fcat^T tile: rows j<64 = f_max, rows j>=64 = relu(Wc*cov9+bc)
  {
    const int mLoc = tid >> 2;          // 0..63
    const int seg  = tid & 3;           // 16-channel segment
    const float* fm = fmax_ws + (size_t)(q0 + mLoc) * CIN + seg * 16;
    float* dst = fcatT + mLoc * 132;
#pragma unroll
    for (int jj = 0; jj < 16; ++jj) dst[seg * 16 + jj] = fm[jj];

    float cv[9];
    const float* cvp = cov9_ws + (size_t)(q0 + mLoc) * 9;
#pragma unroll
    for (int o = 0; o < 9; ++o) cv[o] = cvp[o];
#pragma unroll
    for (int cc = 0; cc < 16; ++cc) {
      const int c = seg * 16 + cc;
      float acc = bc[c];
#pragma unroll
      for (int o = 0; o < 9; ++o) acc += Wc[c * 9 + o] * cv[o];
      dst[64 + c] = fmaxf(acc, 0.0f);
    }
  }
  __syncthreads();

  // ---- WMMA phase
  const int lane   = tid & 31;
  const int wv     = tid >> 5;
  const int ocBase = wv * 16;
  const int nCol   = lane & 15;
  const int hi     = lane >> 4;    // 0: K{0,1}, 1: K{2,3}

  // A fragment source: W_f row (ocBase + nCol), k-offset hi*2
  const float* Arow = Wf + (size_t)(ocBase + nCol) * 128 + hi * 2;
  // B fragment sources: fcatT column n = mb*16 + nCol, k-offset hi*2
  const float* Bp0 = fcatT + (0 * 16 + nCol) * 132 + hi * 2;
  const float* Bp1 = fcatT + (1 * 16 + nCol) * 132 + hi * 2;
  const float* Bp2 = fcatT + (2 * 16 + nCol) * 132 + hi * 2;
  const float* Bp3 = fcatT + (3 * 16 + nCol) * 132 + hi * 2;

  v8f acc[4] = {v8f{}, v8f{}, v8f{}, v8f{}};

#pragma unroll
  for (int k0 = 0; k0 < 128; k0 += 4) {
    const v2f a  = *(const v2f*)(Arow + k0);
    const v2f b0 = *(const v2f*)(Bp0 + k0);
    const v2f b1 = *(const v2f*)(Bp1 + k0);
    const v2f b2 = *(const v2f*)(Bp2 + k0);
    const v2f b3 = *(const v2f*)(Bp3 + k0);
    acc[0] = __builtin_amdgcn_wmma_f32_16x16x4_f32(false, a, false, b0,
                                                   (short)0, acc[0], false, false);
    acc[1] = __builtin_amdgcn_wmma_f32_16x16x4_f32(false, a, false, b1,
                                                   (short)0, acc[1], false, false);
    acc[2] = __builtin_amdgcn_wmma_f32_16x16x4_f32(false, a, false, b2,
                                                   (short)0, acc[2], false, false);
    acc[3] = __builtin_amdgcn_wmma_f32_16x16x4_f32(false, a, false, b3,
                                                   (short)0, acc[3], false, false);
  }

  // ---- bias + relu + store. C/D layout: VGPR v: lanes0-15 row v, lanes16-31 row v+8
#pragma unroll
  for (int mb = 0; mb < 4; ++mb) {
    const int mg = mBase + mb * 16 + nCol;
#pragma unroll
    for (int v = 0; v < 8; ++v) {
      const int oc = ocBase + v + hi * 8;
      const float val = fmaxf(acc[mb][v] + bf[oc], 0.0f);
      out[((size_t)b * COUT + oc) * Mn + mg] = val;
    }
  }
}

// ---------------------------------------------------------------------------
// Launcher
// ---------------------------------------------------------------------------
extern "C" void kernel_launch(void* const* d_in, const int* in_sizes, int n_in,
                              void* d_out, int out_size, void* d_ws, size_t ws_size,
                              hipStream_t stream) {
  const float* f    = (const float*)d_in[0];   // [B, C_IN, N]
  const float* xyz  = (const float*)d_in[1];   // [B, N, 3]
  const float* Wc   = (const float*)d_in[2];   // [C_COV, 9]
  const float* bc   = (const float*)d_in[3];   // [C_COV]
  const float* Wf   = (const float*)d_in[4];   // [C_OUT, 128]
  const float* bf   = (const float*)d_in[5];   // [C_OUT]

  float* out     = (float*)d_out;                       // [B, C_OUT, M]
  float* xyz_new = out + (size_t)Bn * COUT * Mn;        // [B, M, 3]

  // workspace layout
  int*   sidx    = (int*)d_ws;                          // B*M
  int*   knn     = sidx + Bn * Mn;                      // B*M*K
  float* cov9_ws = (float*)(knn + Bn * Mn * Kn);        // B*M*9
  float* fmax_ws = cov9_ws + (size_t)Bn * Mn * 9;       // B*M*C_IN

  fps_kernel  <<<Bn,              256, 0, stream>>>(xyz, sidx, xyz_new);
  knn_kernel  <<<Bn * Mn / 128,   128, 0, stream>>>(xyz, xyz_new, knn);
  group_kernel<<<Bn * Mn / 8,     256, 0, stream>>>(f, xyz, knn, cov9_ws, fmax_ws);
  mlp_kernel  <<<Bn * 16,         256, 0, stream>>>(Wf, bf, Wc, bc, cov9_ws, fmax_ws, out);
}